// PreTrainModel_4355096838991
// MI455X (gfx1250) — compile-verified
//
#include <hip/hip_runtime.h>
#include <hip/hip_bf16.h>
#include <stdint.h>

#define NUME  20000
#define NUMR  10
#define RR    (2 * NUMR)      // 20 directed etypes
#define DIN   200
#define DOUT  200
#define DIMR  200
#define EDGES 500000
#define BB    1024
#define KCAT  (DOUT + DIMR)   // 400

typedef float v2f __attribute__((ext_vector_type(2)));
typedef float v4f __attribute__((ext_vector_type(4)));
typedef float v8f __attribute__((ext_vector_type(8)));

// ---------------------------------------------------------------- utilities

// n4 = count of float4's (all cleared regions are multiples of 4 floats)
__global__ void zero_f32x4(v4f* __restrict__ p, long n4) {
  long i = (long)blockIdx.x * blockDim.x + threadIdx.x;
  long stride = (long)gridDim.x * blockDim.x;
  v4f z = {0.0f, 0.0f, 0.0f, 0.0f};
  for (; i < n4; i += stride) p[i] = z;
}

__global__ void degree_kernel(const int* __restrict__ src,
                              const int* __restrict__ dst,
                              const int* __restrict__ et,
                              float* __restrict__ outdeg,
                              float* __restrict__ indeg) {
  int e = blockIdx.x * blockDim.x + threadIdx.x;
  if (e < EDGES) {
    int t = et[e];
    atomicAdd(&outdeg[t * NUME + src[e]], 1.0f);
    atomicAdd(&indeg [t * NUME + dst[e]], 1.0f);
  }
}

__global__ void norm_kernel(float* __restrict__ p, long n) {
  long i = (long)blockIdx.x * blockDim.x + threadIdx.x;
  if (i < n) {
    float d = p[i];
    p[i] = rsqrtf(d < 1.0f ? 1.0f : d);
  }
}

__global__ void bmean_kernel(const float* __restrict__ b, float* __restrict__ bmean) {
  int d = blockIdx.x * blockDim.x + threadIdx.x;
  if (d < DOUT) {
    float s = 0.0f;
    for (int r = 0; r < RR; ++r) s += b[r * DOUT + d];
    bmean[d] = s * (1.0f / RR);
  }
}

// one wave32 per edge; lanes stripe the feature dim
__global__ void scatter_kernel(const float* __restrict__ h,
                               const float* __restrict__ snorm,
                               const int* __restrict__ src,
                               const int* __restrict__ dst,
                               const int* __restrict__ et,
                               float* __restrict__ agg) {
  int wid  = (blockIdx.x * blockDim.x + threadIdx.x) >> 5;
  int lane = threadIdx.x & 31;
  if (wid >= EDGES) return;
  int s = src[wid], d = dst[wid], t = et[wid];
  float sn = snorm[t * NUME + s];
  const float* hs = h + (size_t)s * DIN;
  float* ag = agg + ((size_t)t * NUME + d) * DIN;
  for (int i = lane; i < DIN; i += 32)
    atomicAdd(&ag[i], hs[i] * sn);
}

// ------------------------------------------------- layer GEMM (WMMA f32 16x16x4)
// hout[n,d] = (1/R) * sum_r [ (dnorm[r,n] * agg[r,n,:]) @ W_r ][n,d] + bmean[d]
// One wave per 16-row strip. Column tiles split across two instantiations
// (NT=7 tiles 0..6, NT=6 tiles 7..12) to stay spill-free. The only partial
// tile (12: cols 192..207 vs DOUT=200) is handled with a clamped address +
// v_cndmask select so no EXEC-mask branches appear in the inner loop.
template <int NT, int T0>
__global__ void layer_gemm_kernel(const float* __restrict__ agg,   // [R,NUME,DIN]
                                  const float* __restrict__ W,     // [R,DIN,DOUT]
                                  const float* __restrict__ dnorm, // [R,NUME]
                                  const float* __restrict__ bmean, // [DOUT]
                                  float* __restrict__ hout) {      // [NUME,DOUT]
  int wave = (blockIdx.x * blockDim.x + threadIdx.x) >> 5;
  int lane = threadIdx.x & 31;
  if (wave * 16 >= NUME) return;
  int row0 = wave * 16;
  int lrow = lane & 15;   // M (for A) / N (for B,C) index within tile
  int hi   = lane >> 4;   // lane-half selects K pair {0,1} vs {2,3}

  v8f acc[NT] = {};

  for (int r = 0; r < RR; ++r) {
    float dn = dnorm[r * NUME + row0 + lrow];
    const float* aggr = agg + ((size_t)r * NUME + row0) * DIN;
    const float* Wr   = W + (size_t)r * DIN * DOUT;
    for (int k0 = 0; k0 < DIN; k0 += 4) {
      int ka = k0 + 2 * hi;
      v2f a;
      a.x = aggr[(size_t)lrow * DIN + ka]     * dn;
      a.y = aggr[(size_t)lrow * DIN + ka + 1] * dn;
      const float* w0 = Wr + (size_t)ka * DOUT;
      const float* w1 = w0 + DOUT;

      float bx[NT], by[NT];
#pragma unroll
      for (int t = 0; t < NT; ++t) {
        int col  = (T0 + t) * 16 + lrow;
        int colc = (col < DOUT) ? col : (DOUT - 1);  // always-valid address
        float vx = w0[colc];                          // unconditional load
        float vy = w1[colc];
        bx[t] = (col < DOUT) ? vx : 0.0f;             // v_cndmask, no branch
        by[t] = (col < DOUT) ? vy : 0.0f;
      }
#pragma unroll
      for (int t = 0; t < NT; ++t) {
        v2f b; b.x = bx[t]; b.y = by[t];
        acc[t] = __builtin_amdgcn_wmma_f32_16x16x4_f32(
            false, a, false, b, (short)0, acc[t], false, false);
      }
    }
  }

#pragma unroll
  for (int t = 0; t < NT; ++t) {
    int col = (T0 + t) * 16 + lrow;
    if (col < DOUT) {
      float bm = bmean[col];
#pragma unroll
      for (int i = 0; i < 8; ++i) {
        int row = row0 + i + 8 * hi;   // C/D layout: VGPR i -> M = i + 8*half
        hout[(size_t)row * DOUT + col] = acc[t][i] * (1.0f / RR) + bm;
      }
    }
  }
}

// ------------------------------------------------- predict GEMM (WMMA f32 16x16x4)
// Y[1024, NUME] = X[1024, 400] @ Wm[400, NUME] + bias
// NUME = 1250 exact 16-col tiles = 156 groups of TN=8 + one TN=2 tail group,
// so NO bounds checks exist in either instantiation: every load/store is
// unconditional and clause-batchable.
template <int TN>
__global__ void predict_gemm_kernel(const float* __restrict__ X,    // [BB,KCAT]
                                    const float* __restrict__ Wm,   // [KCAT,NUME]
                                    const float* __restrict__ bias, // [NUME]
                                    float* __restrict__ Y,          // [BB,NUME]
                                    int colBase) {
  int wib  = threadIdx.x >> 5;
  int lane = threadIdx.x & 31;
  int mstrip = blockIdx.y * 4 + wib;            // 0..63 (exact: BB = 64*16)
  int col0   = colBase + blockIdx.x * (TN * 16);
  int row0   = mstrip * 16;
  int lrow   = lane & 15;
  int hi     = lane >> 4;

  v8f acc[TN] = {};

  const float* xr = X + (size_t)(row0 + lrow) * KCAT;
  for (int k0 = 0; k0 < KCAT; k0 += 4) {
    int ka = k0 + 2 * hi;
    v2f a;
    a.x = xr[ka];
    a.y = xr[ka + 1];
    const float* w0 = Wm + (size_t)ka * NUME + col0 + lrow;
    const float* w1 = w0 + NUME;

    float bx[TN], by[TN];
#pragma unroll
    for (int t = 0; t < TN; ++t) {
      bx[t] = w0[t * 16];
      by[t] = w1[t * 16];
    }
#pragma unroll
    for (int t = 0; t < TN; ++t) {
      v2f b; b.x = bx[t]; b.y = by[t];
      acc[t] = __builtin_amdgcn_wmma_f32_16x16x4_f32(
          false, a, false, b, (short)0, acc[t], false, false);
    }
  }

#pragma unroll
  for (int t = 0; t < TN; ++t) {
    int col = col0 + t * 16 + lrow;
    float bm = bias[col];
#pragma unroll
    for (int i = 0; i < 8; ++i) {
      int row = row0 + i + 8 * hi;
      Y[(size_t)row * NUME + col] = acc[t][i] + bm;
    }
  }
}

// build Xsub = concat(h2[obj], obj_rel[rel]), Xobj = concat(h2[sub], sub_rel[rel])
__global__ void build_x_kernel(const float* __restrict__ h2,
                               const float* __restrict__ sre,
                               const float* __restrict__ ore,
                               const int* __restrict__ sub,
                               const int* __restrict__ obj,
                               const int* __restrict__ rel,
                               float* __restrict__ Xsub,
                               float* __restrict__ Xobj) {
  int i = blockIdx.x;
  int r = rel[i], s = sub[i], o = obj[i];
  for (int d = threadIdx.x; d < KCAT; d += blockDim.x) {
    float vs, vo;
    if (d < DOUT) {
      vs = h2[(size_t)o * DOUT + d];
      vo = h2[(size_t)s * DOUT + d];
    } else {
      vs = ore[(size_t)r * DIMR + (d - DOUT)];
      vo = sre[(size_t)r * DIMR + (d - DOUT)];
    }
    Xsub[(size_t)i * KCAT + d] = vs;
    Xobj[(size_t)i * KCAT + d] = vo;
  }
}

// ---------------------------------------------------------------- launch

extern "C" void kernel_launch(void* const* d_in, const int* in_sizes, int n_in,
                              void* d_out, int out_size, void* d_ws, size_t ws_size,
                              hipStream_t stream) {
  (void)in_sizes; (void)n_in; (void)out_size; (void)ws_size;

  const float* entity_emb = (const float*)d_in[0];
  const float* W0  = (const float*)d_in[1];
  const float* b0  = (const float*)d_in[2];
  const float* W1  = (const float*)d_in[3];
  const float* b1  = (const float*)d_in[4];
  const float* sre = (const float*)d_in[5];   // sub_rel_emb
  const float* ore = (const float*)d_in[6];   // obj_rel_emb
  const float* Wsc = (const float*)d_in[7];
  const float* bsc = (const float*)d_in[8];
  const float* Woc = (const float*)d_in[9];
  const float* boc = (const float*)d_in[10];
  const int* sub   = (const int*)d_in[11];
  const int* obj   = (const int*)d_in[12];
  const int* rel   = (const int*)d_in[13];
  const int* src   = (const int*)d_in[14];
  const int* dst   = (const int*)d_in[15];
  const int* etype = (const int*)d_in[16];

  float* ws    = (float*)d_ws;
  float* snorm = ws;                 // R*NUME = 400000
  float* dnorm = ws + 400000;        // R*NUME
  float* bmean = ws + 800000;        // DOUT
  float* h1    = ws + 1000000;       // NUME*DOUT = 4M
  float* h2    = ws + 5000000;       // NUME*DOUT
  float* Xsub  = ws + 9000000;       // BB*KCAT
  float* Xobj  = ws + 9500000;       // BB*KCAT
  float* agg   = ws + 10000000;      // R*NUME*DIN = 80M floats (320 MB)

  float* sub_pred = (float*)d_out;
  float* obj_pred = sub_pred + (size_t)BB * NUME;

  const long NORM_N = 2L * RR * NUME;          // 800000
  const long AGG_N  = (long)RR * NUME * DIN;   // 80,000,000

  // degrees -> norms (graph identical for both layers)
  zero_f32x4<<<1024, 256, 0, stream>>>((v4f*)snorm, NORM_N / 4);
  degree_kernel<<<(EDGES + 255) / 256, 256, 0, stream>>>(src, dst, etype, snorm, dnorm);
  norm_kernel<<<(int)((NORM_N + 255) / 256), 256, 0, stream>>>(snorm, NORM_N);

  // layer 0: entity_emb -> h1
  bmean_kernel<<<1, 256, 0, stream>>>(b0, bmean);
  zero_f32x4<<<8192, 256, 0, stream>>>((v4f*)agg, AGG_N / 4);
  scatter_kernel<<<(EDGES * 32) / 256, 256, 0, stream>>>(entity_emb, snorm, src, dst, etype, agg);
  layer_gemm_kernel<7, 0><<<(1250 + 3) / 4, 128, 0, stream>>>(agg, W0, dnorm, bmean, h1);
  layer_gemm_kernel<6, 7><<<(1250 + 3) / 4, 128, 0, stream>>>(agg, W0, dnorm, bmean, h1);

  // layer 1: h1 -> h2
  bmean_kernel<<<1, 256, 0, stream>>>(b1, bmean);
  zero_f32x4<<<8192, 256, 0, stream>>>((v4f*)agg, AGG_N / 4);
  scatter_kernel<<<(EDGES * 32) / 256, 256, 0, stream>>>(h1, snorm, src, dst, etype, agg);
  layer_gemm_kernel<7, 0><<<(1250 + 3) / 4, 128, 0, stream>>>(agg, W1, dnorm, bmean, h2);
  layer_gemm_kernel<6, 7><<<(1250 + 3) / 4, 128, 0, stream>>>(agg, W1, dnorm, bmean, h2);

  // prediction
  build_x_kernel<<<BB, 256, 0, stream>>>(h2, sre, ore, sub, obj, rel, Xsub, Xobj);
  dim3 pg_main(156, 16);  // 156 groups of 8 tiles = cols 0..19967
  dim3 pg_tail(1, 16);    // tail: 2 tiles = cols 19968..19999
  predict_gemm_kernel<8><<<pg_main, 128, 0, stream>>>(Xsub, Wsc, bsc, sub_pred, 0);
  predict_gemm_kernel<2><<<pg_tail, 128, 0, stream>>>(Xsub, Wsc, bsc, sub_pred, 19968);
  predict_gemm_kernel<8><<<pg_main, 128, 0, stream>>>(Xobj, Woc, boc, obj_pred, 0);
  predict_gemm_kernel<2><<<pg_tail, 128, 0, stream>>>(Xobj, Woc, boc, obj_pred, 19968);
}